// RGCNBotDetector_87651692577569
// MI455X (gfx1250) — compile-verified
//
#include <hip/hip_runtime.h>
#include <hip/hip_bf16.h>

#define DIM 128

typedef float v2f __attribute__((ext_vector_type(2)));
typedef float v8f __attribute__((ext_vector_type(8)));

// ---------------------------------------------------------------------------
// Fused triple GEMM: xr0 = act(X)@W_rel[0], xr1 = act(X)@W_rel[1],
//                    hroot = act(X)@W_root + bias   (act = optional ReLU)
// fp32 WMMA 16x16x4 (exact fp32, matches fp32 reference).
// grid = ceil(N/128) blocks of 256 threads (8 waves, 16 rows/wave).
// A fragments are loaded to registers ONCE and reused for all 3 matrices;
// weights are staged per-phase into LDS in a K-pair-swizzled layout so each
// B fragment is a single aligned ds_load_b64 (no repacking movs).
// ---------------------------------------------------------------------------
__global__ __launch_bounds__(256)
void rgcn_gemm3_kernel(const float* __restrict__ X,
                       const float* __restrict__ Wrel,   // [2][128][128]
                       const float* __restrict__ Wroot,  // [128][128]
                       const float* __restrict__ bias,   // [128]
                       float* __restrict__ xr0,
                       float* __restrict__ xr1,
                       float* __restrict__ hroot,
                       int n_nodes, int relu_in)
{
    // Swizzled weights: lw[(k>>1)*256 + n*2 + (k&1)] = W[k][n]
    __shared__ float lw[DIM * DIM];   // 64 KB

    const int wave = threadIdx.x >> 5;
    const int lane = threadIdx.x & 31;
    const int row0 = blockIdx.x * 128 + wave * 16;     // 16 rows per wave
    const bool active = (row0 < n_nodes);              // N % 16 == 0

    const int mrow  = lane & 15;          // A: M index
    const int khalf = (lane >> 4) << 1;   // A/B: K sub-offset (0 or 2)
    const int ncol  = lane & 15;          // B/C/D: N index

    // ---- Preload all A fragments (16 rows x 128 K) into registers --------
    const int arow_idx = active ? (row0 + mrow) : 0;
    const float* arow = X + (size_t)arow_idx * DIM + khalf;
    v2f afrag[DIM / 4];
#pragma unroll
    for (int s = 0; s < DIM / 4; ++s) {
        v2f a = *(const v2f*)(arow + s * 4);
        if (relu_in) { a.x = fmaxf(a.x, 0.f); a.y = fmaxf(a.y, 0.f); }
        afrag[s] = a;
    }

    const float* Wm[3] = { Wrel, Wrel + DIM * DIM, Wroot };
    float*       Om[3] = { xr0, xr1, hroot };

    for (int mat = 0; mat < 3; ++mat) {
        __syncthreads();   // previous phase finished reading LDS
        // ---- Stage weight matrix into LDS, K-pair swizzled ---------------
        {
            const float4* src = (const float4*)Wm[mat];
            for (int i = threadIdx.x; i < DIM * DIM / 4; i += 256) {
                float4 v = src[i];
                int k = (i * 4) >> 7;          // row
                int n = (i * 4) & 127;         // col
                float* p = lw + ((k >> 1) << 8) + (k & 1);
                p[(n + 0) * 2] = v.x;
                p[(n + 1) * 2] = v.y;
                p[(n + 2) * 2] = v.z;
                p[(n + 3) * 2] = v.w;
            }
        }
        __syncthreads();

        v8f acc[8];
#pragma unroll
        for (int t = 0; t < 8; ++t)
            acc[t] = (v8f){0.f,0.f,0.f,0.f,0.f,0.f,0.f,0.f};

        // B fragment base: pair (W[k][n], W[k+1][n]) contiguous in LDS.
        const float* bb = lw + ((khalf >> 1) << 8) + ncol * 2;
#pragma unroll
        for (int s = 0; s < DIM / 4; ++s) {      // k0 = 4*s
            const v2f a = afrag[s];
#pragma unroll
            for (int t = 0; t < 8; ++t) {        // column tiles
                v2f b = *(const v2f*)(bb + (s << 9) + (t << 5));
                acc[t] = __builtin_amdgcn_wmma_f32_16x16x4_f32(
                    false, a, false, b, (short)0, acc[t], false, false);
            }
        }

        // C/D layout: VGPR v, lanes 0-15 -> M=v; lanes 16-31 -> M=v+8.
        if (active) {
            float* OUT = Om[mat];
            const int rbase = row0 + ((lane >> 4) << 3);
#pragma unroll
            for (int t = 0; t < 8; ++t) {
                const int col = t * 16 + ncol;
                const float badd = (mat == 2) ? bias[col] : 0.0f;
#pragma unroll
                for (int v = 0; v < 8; ++v)
                    OUT[(size_t)(rbase + v) * DIM + col] = acc[t][v] + badd;
            }
        }
    }
}

// ---------------------------------------------------------------------------
// Edge-count / inverse kernels (counts are layer-invariant).
// ---------------------------------------------------------------------------
__global__ void rgcn_zero_kernel(float* __restrict__ p, int n)
{
    int i = blockIdx.x * blockDim.x + threadIdx.x;
    if (i < n) p[i] = 0.0f;
}

__global__ void rgcn_count_kernel(const int* __restrict__ etype,
                                  const int* __restrict__ dst,
                                  float* __restrict__ cnt, int E, int N)
{
    int e = blockIdx.x * blockDim.x + threadIdx.x;
    if (e < E) atomicAdd(&cnt[etype[e] * N + dst[e]], 1.0f);
}

__global__ void rgcn_inv_kernel(float* __restrict__ cnt, int n)
{
    int i = blockIdx.x * blockDim.x + threadIdx.x;
    if (i < n) cnt[i] = 1.0f / fmaxf(cnt[i], 1.0f);
}

// ---------------------------------------------------------------------------
// Scatter: out[dst] += xr_{type}[src] * inv[type][dst]  (pre-scaled mean).
// 32 lanes per edge, float4 per lane.
// ---------------------------------------------------------------------------
__global__ void rgcn_scatter_kernel(const float* __restrict__ xr0,
                                    const float* __restrict__ xr1,
                                    const float* __restrict__ inv,
                                    const int* __restrict__ src,
                                    const int* __restrict__ dst,
                                    const int* __restrict__ etype,
                                    float* __restrict__ out, int E, int N)
{
    int t = blockIdx.x * blockDim.x + threadIdx.x;
    int e = t >> 5;
    if (e >= E) return;
    int c = (t & 31) * 4;
    int r = etype[e];
    int s = src[e];
    int d = dst[e];
    const float* xr = r ? xr1 : xr0;
    float w = inv[r * N + d];
    float4 v = *(const float4*)(xr + (size_t)s * DIM + c);
    float* o = out + (size_t)d * DIM + c;
    atomicAdd(o + 0, v.x * w);
    atomicAdd(o + 1, v.y * w);
    atomicAdd(o + 2, v.z * w);
    atomicAdd(o + 3, v.w * w);
}

// ---------------------------------------------------------------------------
// Classifier: logits = relu(h) @ Wc + bc  (fuses layer-2 ReLU).
// ---------------------------------------------------------------------------
__global__ void rgcn_cls_kernel(const float* __restrict__ h,
                                const float* __restrict__ Wc,  // [128][2]
                                const float* __restrict__ bc,  // [2]
                                float* __restrict__ out, int n_nodes)
{
    int node = blockIdx.x * blockDim.x + threadIdx.x;
    if (node >= n_nodes) return;
    const float4* hp = (const float4*)(h + (size_t)node * DIM);
    float a0 = 0.f, a1 = 0.f;
#pragma unroll 8
    for (int k4 = 0; k4 < DIM / 4; ++k4) {
        float4 v = hp[k4];
        v.x = fmaxf(v.x, 0.f); v.y = fmaxf(v.y, 0.f);
        v.z = fmaxf(v.z, 0.f); v.w = fmaxf(v.w, 0.f);
        int k = k4 * 4;
        a0 += v.x * Wc[(k+0)*2+0] + v.y * Wc[(k+1)*2+0]
            + v.z * Wc[(k+2)*2+0] + v.w * Wc[(k+3)*2+0];
        a1 += v.x * Wc[(k+0)*2+1] + v.y * Wc[(k+1)*2+1]
            + v.z * Wc[(k+2)*2+1] + v.w * Wc[(k+3)*2+1];
    }
    out[(size_t)node * 2 + 0] = a0 + bc[0];
    out[(size_t)node * 2 + 1] = a1 + bc[1];
}

// ---------------------------------------------------------------------------
extern "C" void kernel_launch(void* const* d_in, const int* in_sizes, int n_in,
                              void* d_out, int out_size, void* d_ws, size_t ws_size,
                              hipStream_t stream)
{
    const float* x       = (const float*)d_in[0];
    const int*   eidx    = (const int*)d_in[1];
    const int*   etype   = (const int*)d_in[2];
    const float* W_rel1  = (const float*)d_in[3];
    const float* W_root1 = (const float*)d_in[4];
    const float* b1      = (const float*)d_in[5];
    const float* W_rel2  = (const float*)d_in[6];
    const float* W_root2 = (const float*)d_in[7];
    const float* b2      = (const float*)d_in[8];
    const float* Wc      = (const float*)d_in[9];
    const float* bc      = (const float*)d_in[10];
    float* out = (float*)d_out;

    const int N = in_sizes[0] / DIM;    // 100000
    const int E = in_sizes[2];          // 500000
    const int* src = eidx;
    const int* dst = eidx + E;

    const size_t ND = (size_t)N * DIM;
    float* ws    = (float*)d_ws;
    float* xr0   = ws;
    float* xr1   = ws + ND;
    float* hbuf  = ws + 2 * ND;
    float* cnt   = ws + 3 * ND;         // 2*N floats

    const int gemm_blocks = (N + 127) / 128;

    // Per-(relation,dst) mean normalization — layer invariant, compute once.
    rgcn_zero_kernel<<<(2 * N + 255) / 256, 256, 0, stream>>>(cnt, 2 * N);
    rgcn_count_kernel<<<(E + 255) / 256, 256, 0, stream>>>(etype, dst, cnt, E, N);
    rgcn_inv_kernel<<<(2 * N + 255) / 256, 256, 0, stream>>>(cnt, 2 * N);

    // Layer 1: h = x@W_root1 + b1 (+ scatter of x@W_rel1[r])
    rgcn_gemm3_kernel<<<gemm_blocks, 256, 0, stream>>>(
        x, W_rel1, W_root1, b1, xr0, xr1, hbuf, N, 0);
    rgcn_scatter_kernel<<<(E * 32 + 255) / 256, 256, 0, stream>>>(
        xr0, xr1, cnt, src, dst, etype, hbuf, E, N);

    // Layer 2: ReLU fused into GEMM A-fragment load; in-place hbuf -> hbuf.
    rgcn_gemm3_kernel<<<gemm_blocks, 256, 0, stream>>>(
        hbuf, W_rel2, W_root2, b2, xr0, xr1, hbuf, N, 1);
    rgcn_scatter_kernel<<<(E * 32 + 255) / 256, 256, 0, stream>>>(
        xr0, xr1, cnt, src, dst, etype, hbuf, E, N);

    // Classifier (fuses layer-2 ReLU on read)
    rgcn_cls_kernel<<<(N + 255) / 256, 256, 0, stream>>>(hbuf, Wc, bc, out, N);
}